// QModelGen595_65481071400327
// MI455X (gfx1250) — compile-verified
//
#include <hip/hip_runtime.h>

typedef float v2f __attribute__((ext_vector_type(2)));
typedef float v4f __attribute__((ext_vector_type(4)));
typedef float v8f __attribute__((ext_vector_type(8)));
typedef int   v4i __attribute__((vector_size(16)));

typedef __attribute__((address_space(1))) v4i* gv4i_ptr;   // global (AS1)
typedef __attribute__((address_space(3))) v4i* lv4i_ptr;   // LDS (AS3)

#define NW   16
#define DIM  65536      // 2^16 amplitudes
#define TPB  256        // 8 wave32

#if defined(__has_builtin)
#if __has_builtin(__builtin_amdgcn_global_load_async_to_lds_b128)
#define HAVE_ASYNC_LDS 1
#endif
#endif

// Entry of the 16x16 tensor-product rotation G = R(t0) (x) R(t1) (x) R(t2) (x) R(t3).
// Field bit (3-j) of the 4-bit index corresponds to wire (4g+j).
// R = [[c,-s],[s,c]]:  a==b -> c ; a=0,b=1 -> -s ; a=1,b=0 -> s
__device__ __forceinline__ float gentry(int a, int b, const float c4[4], const float s4[4]) {
  float r = 1.0f;
#pragma unroll
  for (int j = 0; j < 4; ++j) {
    int bp = 3 - j;
    int aj = (a >> bp) & 1;
    int bj = (b >> bp) & 1;
    float f = (aj == bj) ? c4[j] : (aj ? s4[j] : -s4[j]);
    r *= f;
  }
  return r;
}

__global__ __launch_bounds__(TPB) void qsim_kernel(const float* __restrict__ state,
                                                   const float* __restrict__ theta,
                                                   const float* __restrict__ head_w,
                                                   const float* __restrict__ head_b,
                                                   float* __restrict__ out) {
  extern __shared__ float lds[];   // 65536 floats = 256 KB (<= 320 KB WGP LDS)
  const int tid  = (int)threadIdx.x;
  const int bidx = (int)blockIdx.x;
  const int lane = tid & 31;
  const int wave = tid >> 5;

  // ---- stage one batch element's state into LDS ----
  // Preferred: CDNA5 async global->LDS (no VGPR round-trip; ASYNCcnt-tracked).
  {
    const v4f* src = (const v4f*)(state + (size_t)bidx * DIM);
#if defined(HAVE_ASYNC_LDS)
    v4f* dstg = (v4f*)lds;
#pragma unroll 4
    for (int e = tid; e < DIM / 4; e += TPB) {
      __builtin_amdgcn_global_load_async_to_lds_b128(
          (gv4i_ptr)(src + e),      // global source (AS1, v4i pointee)
          (lv4i_ptr)(dstg + e),     // LDS destination (AS3, v4i pointee)
          /*imm offset*/ 0, /*cpol*/ 0);
    }
#if __has_builtin(__builtin_amdgcn_s_wait_asynccnt)
    __builtin_amdgcn_s_wait_asynccnt(0);
#else
    asm volatile("s_wait_asynccnt 0x0" ::: "memory");
#endif
#else
    v4f* dst = (v4f*)lds;
#pragma unroll 4
    for (int e = tid; e < DIM / 4; e += TPB) {
      dst[e] = __builtin_nontemporal_load(&src[e]);
    }
#endif
  }
  __syncthreads();

  const int row   = lane & 15;   // WMMA M-row (A) / N-col (B,D) for this lane
  const int khalf = lane >> 4;   // upper half-wave holds the upper half of K / M

  // ---- 4 groups of 4 wires; each is a 16x16 GEMM along one 4-bit axis, in-place in LDS ----
  for (int g = 0; g < 4; ++g) {
    const int p = 12 - 4 * g;            // field bit position of this group's axis
    float c4[4], s4[4];
#pragma unroll
    for (int j = 0; j < 4; ++j) {
      float th = 0.5f * theta[4 * g + j];
      c4[j] = __builtin_cosf(th);
      s4[j] = __builtin_sinf(th);
    }
    // A fragments (16x4 f32 per chunk): lane holds A[row][4*kc + 2*khalf + {0,1}]
    v2f afrag[4];
#pragma unroll
    for (int kc = 0; kc < 4; ++kc) {
      int k0 = 4 * kc + 2 * khalf;
      afrag[kc][0] = gentry(row, k0,     c4, s4);
      afrag[kc][1] = gentry(row, k0 + 1, c4, s4);
    }

    const int pm = (1 << p) - 1;
    // 4096 columns (the other 12 bits), 16 per tile -> 256 tiles over 8 waves
    for (int tile = wave; tile < 256; tile += TPB / 32) {
      int o     = tile * 16 + row;                       // this lane's column index
      int obase = ((o >> p) << (p + 4)) | (o & pm);      // insert 4-bit field at bit p

      v8f acc = {0.f, 0.f, 0.f, 0.f, 0.f, 0.f, 0.f, 0.f};
#pragma unroll
      for (int kc = 0; kc < 4; ++kc) {
        int kb = 4 * kc + 2 * khalf;
        v2f bfrag;
        bfrag[0] = lds[obase + (kb << p)];
        bfrag[1] = lds[obase + ((kb + 1) << p)];
        acc = __builtin_amdgcn_wmma_f32_16x16x4_f32(false, afrag[kc], false, bfrag,
                                                    (short)0, acc, false, false);
      }
      // D layout: lanes 0-15 rows 0..7, lanes 16-31 rows 8..15, same column
#pragma unroll
      for (int r = 0; r < 8; ++r) {
        lds[obase + ((r + 8 * khalf) << p)] = acc[r];
      }
    }
    __syncthreads();
  }

  // ---- reduction: out = head_b + sum_i psi[i]^2 * S(i),
  // S(i) = sum_w head_w[w] * (-1)^{prefix parity of top (w+1) bits of i}.
  // With i = e*256 + tid: wires 0..7 depend only on e, wires 8..15 only on tid.
  float hw[NW];
#pragma unroll
  for (int w = 0; w < NW; ++w) hw[w] = head_w[w];

  float slo = 0.f;
  {
    int par = 0;
#pragma unroll
    for (int w = 8; w < 16; ++w) {
      par ^= (tid >> (15 - w)) & 1;
      slo += hw[w] * (par ? -1.f : 1.f);
    }
  }

  float acc1 = 0.f, acc2 = 0.f;   // sum v^2*Shi(e)  and  sum v^2*par8(e)
  for (int e = 0; e < DIM / TPB; ++e) {
    float v  = lds[e * TPB + tid];
    float pr = v * v;
    int par = 0; float shi = 0.f;
#pragma unroll
    for (int w = 0; w < 8; ++w) {
      par ^= (e >> (7 - w)) & 1;
      shi += hw[w] * (par ? -1.f : 1.f);
    }
    acc1 = __builtin_fmaf(pr, shi, acc1);
    acc2 = par ? (acc2 - pr) : (acc2 + pr);
  }
  float acc = acc1 + slo * acc2;

  __syncthreads();                 // everyone done reading state; reuse LDS front
  lds[tid] = acc;
  __syncthreads();
  for (int s = TPB / 2; s > 0; s >>= 1) {
    if (tid < s) lds[tid] += lds[tid + s];
    __syncthreads();
  }
  if (tid == 0) out[bidx] = lds[0] + head_b[0];
}

extern "C" void kernel_launch(void* const* d_in, const int* in_sizes, int n_in,
                              void* d_out, int out_size, void* d_ws, size_t ws_size,
                              hipStream_t stream) {
  const float* state  = (const float*)d_in[0];
  const float* theta  = (const float*)d_in[1];
  // d_in[2] = phi: unused — RZ phases have unit modulus and cancel in |psi|^2.
  const float* head_w = (const float*)d_in[3];
  const float* head_b = (const float*)d_in[4];
  float* out = (float*)d_out;

  const int B = in_sizes[0] / DIM;   // 512
  qsim_kernel<<<B, TPB, DIM * sizeof(float), stream>>>(state, theta, head_w, head_b, out);
}